// HGTLayerSingle_78142634983559
// MI455X (gfx1250) — compile-verified
//
#include <hip/hip_runtime.h>
#include <hip/hip_bf16.h>
#include <math.h>

// MI455X / gfx1250: wave32, WMMA (not MFMA). fp32 path via V_WMMA_F32_16X16X4_F32.

typedef float v2f __attribute__((ext_vector_type(2)));
typedef float v8f __attribute__((ext_vector_type(8)));

#define DIM 128            // IN_DIM == OUT_DIM == 128
#define NHEAD 4
#define HEAD_DIM 32
#define MAXN 32
#define LN_EPS 1e-5f

// ---------------- wave32 butterfly reductions ----------------
__device__ __forceinline__ float wave_sum32(float v) {
#pragma unroll
  for (int m = 16; m >= 1; m >>= 1) v += __shfl_xor(v, m, 32);
  return v;
}
__device__ __forceinline__ float wave_max32(float v) {
#pragma unroll
  for (int m = 16; m >= 1; m >>= 1) v = fmaxf(v, __shfl_xor(v, m, 32));
  return v;
}

// ---------------- fp32 WMMA 16x16 tile: D = A[r0:r0+16, 0:128] * W[0:128, c0:c0+16]
// A-frag (ISA 7.12.2, 32-bit A 16x4): lane l<16 holds row M=l, VGPR0=K0,VGPR1=K1;
// lanes 16..31 hold K2/K3.  B rows striped across lanes: lanes<16 rows k+0/k+1,
// lanes>=16 rows k+2/k+3, col = lane&15.  C/D: VGPR v -> rows v (lanes<16), v+8.
__device__ __forceinline__ v8f wmma_f32_tile(const float* __restrict__ A,
                                             const float* __restrict__ W,
                                             int r0, int c0, int lane, int n) {
  const int half = lane >> 4;
  const int l    = lane & 15;
  int row = r0 + l; if (row > n - 1) row = n - 1;   // clamp loads, keep EXEC full
  const float2* a2 = (const float2*)(A + (size_t)row * DIM);
  const float*  wc = W + c0 + l;
  v8f acc = {0.f,0.f,0.f,0.f,0.f,0.f,0.f,0.f};
#pragma unroll
  for (int k = 0; k < DIM; k += 4) {
    float2 av = a2[(k >> 1) + half];               // A[row][k+2*half], [..+1]
    v2f a; a.x = av.x; a.y = av.y;
    v2f b;
    b.x = wc[(size_t)(k + 2 * half)     * DIM];    // W[k+2h  ][c0+l]
    b.y = wc[(size_t)(k + 2 * half + 1) * DIM];    // W[k+2h+1][c0+l]
    acc = __builtin_amdgcn_wmma_f32_16x16x4_f32(false, a, false, b,
                                                (short)0, acc, false, false);
  }
  return acc;
}

__device__ __forceinline__ void store_tile(float* __restrict__ O, v8f acc,
                                           int r0, int c0, int lane, int n) {
  const int half = lane >> 4, l = lane & 15;
#pragma unroll
  for (int v = 0; v < 8; ++v) {
    int row = r0 + v + 8 * half;
    if (row < n) O[(size_t)row * DIM + c0 + l] = acc[v];
  }
}

// ---------------- Kernel A: Q/K/V = h @ {Wq,Wk,Wv} ----------------
__global__ __launch_bounds__(256) void qkv_gemm_kernel(
    const float* __restrict__ h,
    const float* __restrict__ Wq, const float* __restrict__ Wk,
    const float* __restrict__ Wv,
    float* __restrict__ Q, float* __restrict__ K, float* __restrict__ V, int n) {
  const int wave = threadIdx.x >> 5, lane = threadIdx.x & 31;
  const int r0 = blockIdx.x * 16, c0 = wave * 16;
  const float* W = (blockIdx.y == 0) ? Wq : (blockIdx.y == 1) ? Wk : Wv;
  float*       O = (blockIdx.y == 0) ? Q  : (blockIdx.y == 1) ? K  : V;
  v8f acc = wmma_f32_tile(h, W, r0, c0, lane, n);
  store_tile(O, acc, r0, c0, lane, n);
}

// ---------------- Kernel B: gather + masked softmax attention --------------
// One wave per node; lane = neighbor for score phase, lane = dim for V phase.
__global__ __launch_bounds__(256) void attn_kernel(
    const float* __restrict__ Q, const float* __restrict__ K,
    const float* __restrict__ V,
    const long long* __restrict__ nidx, const int* __restrict__ nmask,
    float* __restrict__ OH, int n) {
  __shared__ float s_q[8][DIM];
  __shared__ float s_alpha[8][MAXN][NHEAD];
  __shared__ int   s_idx[8][MAXN];

  const int wave = threadIdx.x >> 5, lane = threadIdx.x & 31;
  int node = blockIdx.x * 8 + wave;
  const bool active = node < n;
  if (node > n - 1) node = n - 1;

  // stage this node's Q row and neighbor list
#pragma unroll
  for (int j = 0; j < 4; ++j)
    s_q[wave][lane + 32 * j] = Q[(size_t)node * DIM + lane + 32 * j];
  const long long idx = nidx[(size_t)node * MAXN + lane];
  const int       msk = nmask[(size_t)node * MAXN + lane];
  s_idx[wave][lane] = (int)idx;
  __syncthreads();

  // per-lane (= per-neighbor) scores for all 4 heads
  const float4* k4 = (const float4*)(K + (size_t)idx * DIM);
  const float scale = 0.17677669529663687f;   // 1/sqrt(32)
  float sc[NHEAD];
#pragma unroll
  for (int hh = 0; hh < NHEAD; ++hh) {
    float acc = 0.f;
#pragma unroll
    for (int j = 0; j < 8; ++j) {
      float4 kv = k4[hh * 8 + j];
      int b = hh * HEAD_DIM + j * 4;
      acc += s_q[wave][b + 0] * kv.x + s_q[wave][b + 1] * kv.y +
             s_q[wave][b + 2] * kv.z + s_q[wave][b + 3] * kv.w;
    }
    sc[hh] = msk ? acc * scale : -1.0e9f;
  }

  // softmax across the 32 neighbors (cross-lane, wave32)
#pragma unroll
  for (int hh = 0; hh < NHEAD; ++hh) {
    float mx = wave_max32(sc[hh]);
    float e  = expf(sc[hh] - mx);
    float sm = wave_sum32(e);
    s_alpha[wave][lane][hh] = e / sm;
  }
  __syncthreads();

  // lane = intra-head dim; accumulate alpha-weighted V (coalesced per-m reads)
#pragma unroll
  for (int c = 0; c < NHEAD; ++c) {
    float acc = 0.f;
    for (int m = 0; m < MAXN; ++m) {
      acc += s_alpha[wave][m][c] *
             V[(size_t)s_idx[wave][m] * DIM + c * HEAD_DIM + lane];
    }
    if (active) OH[(size_t)node * DIM + c * HEAD_DIM + lane] = acc;
  }
}

// ---------------- Kernel C: fc + bias + residual + GELU + LayerNorm --------
__global__ __launch_bounds__(256) void fc_gelu_ln_kernel(
    const float* __restrict__ OH, const float* __restrict__ Wfc,
    const float* __restrict__ bfc, const float* __restrict__ Q,
    const float* __restrict__ gamma, const float* __restrict__ beta,
    float* __restrict__ out, int n) {
  __shared__ float tile[16][132];   // pad: rows v and v+8 land on distinct banks
  const int wave = threadIdx.x >> 5, lane = threadIdx.x & 31;
  const int r0 = blockIdx.x * 16, c0 = wave * 16;

  v8f acc = wmma_f32_tile(OH, Wfc, r0, c0, lane, n);
  const int half = lane >> 4, l = lane & 15;
#pragma unroll
  for (int v = 0; v < 8; ++v) tile[v + 8 * half][c0 + l] = acc[v];
  __syncthreads();

  // bias + residual(Q) + exact GELU
  for (int i = threadIdx.x; i < 16 * DIM; i += 256) {
    int r = i >> 7, c = i & 127;
    int row = r0 + r; if (row > n - 1) row = n - 1;
    float x = tile[r][c] + bfc[c] + Q[(size_t)row * DIM + c];
    x = 0.5f * x * (1.0f + erff(x * 0.70710678118654752f));
    tile[r][c] = x;
  }
  __syncthreads();

  // LayerNorm: each wave handles 2 rows, 4 elems/lane
#pragma unroll
  for (int rr = 0; rr < 2; ++rr) {
    int r = wave * 2 + rr;
    float x0 = tile[r][lane],      x1 = tile[r][lane + 32];
    float x2 = tile[r][lane + 64], x3 = tile[r][lane + 96];
    float mean = wave_sum32(x0 + x1 + x2 + x3) * (1.0f / 128.0f);
    float d0 = x0 - mean, d1 = x1 - mean, d2 = x2 - mean, d3 = x3 - mean;
    float var = wave_sum32(d0*d0 + d1*d1 + d2*d2 + d3*d3) * (1.0f / 128.0f);
    float inv = rsqrtf(var + LN_EPS);
    int row = r0 + r;
    if (row < n) {
      out[(size_t)row * DIM + lane]      = d0 * inv * gamma[lane]      + beta[lane];
      out[(size_t)row * DIM + lane + 32] = d1 * inv * gamma[lane + 32] + beta[lane + 32];
      out[(size_t)row * DIM + lane + 64] = d2 * inv * gamma[lane + 64] + beta[lane + 64];
      out[(size_t)row * DIM + lane + 96] = d3 * inv * gamma[lane + 96] + beta[lane + 96];
    }
  }
}

// ---------------- launch ----------------
extern "C" void kernel_launch(void* const* d_in, const int* in_sizes, int n_in,
                              void* d_out, int out_size, void* d_ws, size_t ws_size,
                              hipStream_t stream) {
  const float*     h     = (const float*)d_in[0];
  const float*     Wq    = (const float*)d_in[1];
  const float*     Wk    = (const float*)d_in[2];
  const float*     Wv    = (const float*)d_in[3];
  const float*     Wfc   = (const float*)d_in[4];
  const float*     bfc   = (const float*)d_in[5];
  const float*     gamma = (const float*)d_in[6];
  const float*     beta  = (const float*)d_in[7];
  const long long* nidx  = (const long long*)d_in[8];   // int64 in reference
  const int*       nmask = (const int*)d_in[9];
  float* out = (float*)d_out;

  const int n = in_sizes[0] / DIM;          // 30000
  const size_t per = (size_t)n * DIM;       // floats per [N,128] buffer
  float* Q  = (float*)d_ws;                 // 15.36 MB each; 61.4 MB total in ws
  float* K  = Q + per;
  float* V  = K + per;
  float* OH = V + per;

  const int nb = (n + 15) / 16;             // 1875 row blocks
  qkv_gemm_kernel<<<dim3(nb, 3), 256, 0, stream>>>(h, Wq, Wk, Wv, Q, K, V, n);
  attn_kernel<<<dim3((n + 7) / 8), 256, 0, stream>>>(Q, K, V, nidx, nmask, OH, n);
  fc_gelu_ln_kernel<<<dim3(nb), 256, 0, stream>>>(OH, Wfc, bfc, Q, gamma, beta, out, n);
}